// GlobalCrossDecode_60129542794
// MI455X (gfx1250) — compile-verified
//
#include <hip/hip_runtime.h>
#include <hip/hip_bf16.h>

typedef __attribute__((ext_vector_type(2))) float v2f;
typedef __attribute__((ext_vector_type(4))) float v4f;
typedef __attribute__((ext_vector_type(8))) float v8f;

#define B_    8
#define L_    4096
#define DTOK  1024
#define DLAT  512

// ---------------------------------------------------------------------------
// Kernel 1: y[b,:] = Wout @ (Wkv_v @ kv[b] + bkv_v) + bout for b=0..7, via
// V_WMMA_F32_16X16X4_F32 (full fp32, matches reference math exactly since
// softmax over a single key is identically 1 and the q path is dead).
//
// M is padded 8 -> 16 by *duplicating* rows (mn & 7): a row of A only affects
// the same row of D, and D rows 8..15 (lanes 16..31 of the C fragment) are
// never stored, so duplication is safe and keeps every load unconditional.
//
// Fragment layouts (CDNA5 ISA 7.12.2, 32-bit 16x4 A / 4x16 B / 16x16 C):
//   A: lane<16 -> (M=lane, K=k0+{0,1}) in v0/v1; lane>=16 -> K=k0+{2,3}.
//   B: lane<16 -> (K=k0+{0,1}, N=lane); lane>=16 -> K=k0+{2,3}.
//   C: VGPR r, lanes 0..15 -> (M=r, N=lane)  => lanes 0..15 hold batches 0..7.
// ---------------------------------------------------------------------------
__global__ void __launch_bounds__(1024)
mla_decode_y_kernel(const float* __restrict__ kv,    // (8, 512)
                    const float* __restrict__ Wkv,   // (2048, 512) row-major
                    const float* __restrict__ bkv,   // (2048)
                    const float* __restrict__ Wout,  // (1024, 1024) row-major
                    const float* __restrict__ bout,  // (1024)
                    float* __restrict__ y)           // (8, 1024) workspace
{
    __shared__ float tmp[B_][DTOK];   // v-projection, 32 KB of 320 KB LDS

    const int tid  = threadIdx.x;
    const int wave = tid >> 5;
    const int lane = tid & 31;
    const int mn   = lane & 15;       // A-row (M) / B-col (N) index
    const int m8   = mn & 7;          // clamped batch row (pad rows duplicate)
    const int kh   = lane >> 4;       // which K pair this half-wave supplies

    // ---- Phase A: tmp[m][n] = sum_k kv[m][k] * Wkv[1024+n][k] + bkv[1024+n]
    const float* arow = kv + (size_t)m8 * DLAT;
    for (int t = 0; t < 2; ++t) {
        const int n0 = (wave + t * 32) * 16;
        v8f c = {};
        const float* wrow = Wkv + (size_t)(DTOK + n0 + mn) * DLAT;
#pragma unroll 4
        for (int k0 = 0; k0 < DLAT; k0 += 4) {
            const int ka = k0 + 2 * kh;
            const v2f a = *(const v2f*)(arow + ka);   // global_load_b64
            const v2f b = *(const v2f*)(wrow + ka);   // global_load_b64
            c = __builtin_amdgcn_wmma_f32_16x16x4_f32(
                    false, a, false, b, (short)0, c, false, false);
        }
        if (lane < 16) {
            const float bias = bkv[DTOK + n0 + mn];
#pragma unroll
            for (int r = 0; r < B_; ++r)
                tmp[r][n0 + mn] = c[r] + bias;
        }
    }
    __syncthreads();

    // ---- Phase B: y[m][n] = sum_k tmp[m][k] * Wout[n][k] + bout[n]
    for (int t = 0; t < 2; ++t) {
        const int n0 = (wave + t * 32) * 16;
        v8f c = {};
        const float* wrow = Wout + (size_t)(n0 + mn) * DTOK;
#pragma unroll 4
        for (int k0 = 0; k0 < DTOK; k0 += 4) {
            const int ka = k0 + 2 * kh;
            const v2f a = *(const v2f*)(&tmp[m8][ka]); // ds_load_b64
            const v2f b = *(const v2f*)(wrow + ka);    // global_load_b64
            c = __builtin_amdgcn_wmma_f32_16x16x4_f32(
                    false, a, false, b, (short)0, c, false, false);
        }
        if (lane < 16) {
            const float bias = bout[n0 + mn];
#pragma unroll
            for (int r = 0; r < B_; ++r)
                y[(size_t)r * DTOK + n0 + mn] = c[r] + bias;
        }
    }
}

// ---------------------------------------------------------------------------
// Kernel 2: out[b, l, :] = y[b, :] for all l. Pure store-bandwidth kernel:
// 134 MB of float4 stores (fits in the 192 MB L2); the 4 KB y row is an L2
// hit for every block. This kernel is the entire runtime (~6 us roofline).
// ---------------------------------------------------------------------------
__global__ void __launch_bounds__(256)
mla_decode_broadcast_kernel(const float* __restrict__ y,   // (8, 1024)
                            float* __restrict__ out)       // (8, 4096, 1024)
{
    const unsigned row = blockIdx.x;            // 0 .. 8*4096-1
    const unsigned b   = row >> 12;             // row / 4096
    const v4f* __restrict__ src = (const v4f*)(y + (size_t)b * DTOK);
    v4f* __restrict__ dst       = (v4f*)(out + (size_t)row * DTOK);
    dst[threadIdx.x] = src[threadIdx.x];        // 256 lanes x 16 B = full row
}

extern "C" void kernel_launch(void* const* d_in, const int* in_sizes, int n_in,
                              void* d_out, int out_size, void* d_ws, size_t ws_size,
                              hipStream_t stream) {
    // setup_inputs order: q, kv, Wq, bq, Wkv, bkv, Wout, bout (all fp32).
    // q / Wq / bq are mathematically dead (softmax over a single key == 1).
    const float* kv   = (const float*)d_in[1];
    const float* Wkv  = (const float*)d_in[4];
    const float* bkv  = (const float*)d_in[5];
    const float* Wout = (const float*)d_in[6];
    const float* bout = (const float*)d_in[7];

    float* y   = (float*)d_ws;     // 8 * 1024 * 4 B = 32 KB scratch
    float* out = (float*)d_out;

    mla_decode_y_kernel<<<1, 1024, 0, stream>>>(kv, Wkv, bkv, Wout, bout, y);
    mla_decode_broadcast_kernel<<<B_ * L_, 256, 0, stream>>>(y, out);
}